// TextureDifference_8821862826415
// MI455X (gfx1250) — compile-verified
//
#include <hip/hip_runtime.h>
#include <stdint.h>

#define IMG_H 512
#define IMG_W 512
#define IMG_B 16
#define TW 128
#define TH 16
#define HALO 2
#define GW (TW + 2*HALO)   /* 132 */
#define GH (TH + 2*HALO)   /* 20  */
#define NT 256
#define GAREA (GH*GW)      /* 2640 */
#define HAREA (GH*TW)      /* 2560 */
#define OAREA (TH*TW)      /* 2048 */
#define PER_T (OAREA/NT)   /* 8 */

typedef __attribute__((address_space(1))) int  g_i32;
typedef __attribute__((address_space(3))) int  l_i32;

// ---- CDNA5 async global->LDS copy (ASYNCcnt path), with safe fallback ----
__device__ __forceinline__ void async_g2l(const float* g, float* l) {
#if __has_builtin(__builtin_amdgcn_global_load_async_to_lds_b32)
    __builtin_amdgcn_global_load_async_to_lds_b32(
        (g_i32*)(uintptr_t)g,   // AS1 and generic share addresses for global mem
        (l_i32*)l,              // addrspacecast generic -> LDS
        0, 0);
#else
    *l = *g;
#endif
}

__device__ __forceinline__ void wait_async_all() {
#if __has_builtin(__builtin_amdgcn_s_wait_asynccnt)
    __builtin_amdgcn_s_wait_asynccnt(0);
#else
    asm volatile("s_wait_asynccnt 0x0" ::: "memory");
#endif
}

__device__ __forceinline__ int reflect_idx(int i, int n) {
    i = (i < 0) ? -i : i;              // -1 -> 1, -2 -> 2 (jnp reflect)
    i = (i >= n) ? (2*n - 2 - i) : i;  // n -> n-2, n+1 -> n-3
    return i;
}

__global__ __launch_bounds__(NT)
void texdiff_kernel(const float* __restrict__ img1,
                    const float* __restrict__ img2,
                    float* __restrict__ out)
{
    // 42,240 B of LDS; h-buffers alias channel buffers (phases separated by barriers)
    __shared__ float smem[4 * GAREA];
    float* const s_g  = smem;              // gray tile           [GH][GW]
    float* const s_c0 = smem + 1*GAREA;    // raw ch0             [GH][GW]
    float* const s_c1 = smem + 2*GAREA;    // raw ch1             [GH][GW]
    float* const s_c2 = smem + 3*GAREA;    // raw ch2             [GH][GW]
    float* const s_h1 = smem + 1*GAREA;    // horiz sum(g)        [GH][TW] (aliases c0)
    float* const s_h2 = smem + 2*GAREA;    // horiz sum(g^2)      [GH][TW] (aliases c1)

    const int tid = threadIdx.x;
    const int x0  = blockIdx.x * TW;
    const int y0  = blockIdx.y * TH;
    const int b   = blockIdx.z;
    const size_t plane = (size_t)IMG_H * IMG_W;

    float sdev[2][PER_T];

    for (int im = 0; im < 2; ++im) {
        const float* img = (im == 0 ? img1 : img2) + (size_t)b * 3 * plane;

        // ---- Phase 1: async-stage 3 raw channel tiles (halo reflected at fill) ----
        for (int t = tid; t < GAREA; t += NT) {
            const int ly = t / GW;
            const int lx = t - ly * GW;
            const int gy = reflect_idx(y0 - HALO + ly, IMG_H);
            const int gx = reflect_idx(x0 - HALO + lx, IMG_W);
            const float* p = img + (size_t)gy * IMG_W + gx;
            async_g2l(p,             &s_c0[t]);
            async_g2l(p + plane,     &s_c1[t]);
            async_g2l(p + 2*plane,   &s_c2[t]);
        }
        wait_async_all();
        __syncthreads();

        // ---- Phase 2: gray = 0.144*c0 + 0.587*c1 + 0.299*c2 ----
        for (int t = tid; t < GAREA; t += NT) {
            s_g[t] = 0.144f * s_c0[t] + 0.587f * s_c1[t] + 0.299f * s_c2[t];
        }
        __syncthreads();

        // ---- Phase 3: horizontal 5-tap sums of g and g^2 ----
        for (int t = tid; t < HAREA; t += NT) {
            const int ly = t >> 7;          // TW == 128
            const int tx = t & (TW - 1);
            const float* r = &s_g[ly * GW + tx];
            const float a = r[0], b2 = r[1], c = r[2], d = r[3], e = r[4];
            s_h1[t] = a + b2 + c + d + e;
            s_h2[t] = a*a + b2*b2 + c*c + d*d + e*e;
        }
        __syncthreads();

        // ---- Phase 4: vertical 5-tap sums -> local stddev (registers) ----
        #pragma unroll
        for (int k = 0; k < PER_T; ++k) {
            const int t  = tid + k * NT;
            const int ty = t >> 7;
            const int tx = t & (TW - 1);
            float sum = 0.f, sum2 = 0.f;
            #pragma unroll
            for (int dy = 0; dy < 5; ++dy) {
                sum  += s_h1[(ty + dy) * TW + tx];
                sum2 += s_h2[(ty + dy) * TW + tx];
            }
            const float m   = sum  / 25.0f;
            const float ex2 = sum2 / 25.0f;
            const float var = fmaxf(ex2 - m * m, 0.0f);
            sdev[im][k] = __builtin_sqrtf(var + 1e-9f);
        }
        __syncthreads();  // protect h-buffers before next image's fill reuses LDS
    }

    // ---- Combine + threshold + streaming store ----
    #pragma unroll
    for (int k = 0; k < PER_T; ++k) {
        const int t  = tid + k * NT;
        const int ty = t >> 7;
        const int tx = t & (TW - 1);
        const float s1 = sdev[0][k];
        const float s2 = sdev[1][k];
        const float diff = (2.0f * s1 * s2) / (s1*s1 + s2*s2 + 1e-5f + 1e-8f);
        const float v = (diff > 0.975f) ? 1.0f : 0.0f;
        __builtin_nontemporal_store(v, &out[(size_t)b * plane + (size_t)(y0 + ty) * IMG_W + (x0 + tx)]);
    }
}

extern "C" void kernel_launch(void* const* d_in, const int* in_sizes, int n_in,
                              void* d_out, int out_size, void* d_ws, size_t ws_size,
                              hipStream_t stream) {
    const float* img1 = (const float*)d_in[0];
    const float* img2 = (const float*)d_in[1];
    float* out = (float*)d_out;
    dim3 grid(IMG_W / TW, IMG_H / TH, IMG_B);  // 4 x 32 x 16
    texdiff_kernel<<<grid, NT, 0, stream>>>(img1, img2, out);
}